// CollaborativeFusion_DeCooper_39152921870881
// MI455X (gfx1250) — compile-verified
//
#include <hip/hip_runtime.h>

// CDNA5 / gfx1250: wave32, WMMA f32 16x16x4 used for the exact fp32 channel
// reduction (imp = |A| x ones). Bandwidth-bound op: ~300MB traffic, ~13us at
// 23.3 TB/s; inputs (100MB) fit in 192MB L2 so the apply pass re-reads cheap.

typedef float v2f __attribute__((ext_vector_type(2)));
typedef float v8f __attribute__((ext_vector_type(8)));

#define HW_   4096   // 64*64
#define C_    256
#define B_    8
#define K_SEL 2048   // 0.5 * HW

// ---------------------------------------------------------------------------
// Pass 1: imp[t][b][hw] = sum_c |fm[t][b][c][hw]|  via V_WMMA_F32_16X16X4_F32.
// One wave per (t, b, 16-pixel tile): A = 16(hw) x 4(c) of |fm|, B = ones,
// accumulate f32 over 64 K-steps. D row m (any column) = channel sum of pixel
// hw_base+m; lane 0 holds rows 0..7 (N=0), lane 16 holds rows 8..15.
// ---------------------------------------------------------------------------
__global__ void __launch_bounds__(256)
importance_wmma_kernel(const float* __restrict__ in0,
                       const float* __restrict__ in1,
                       const float* __restrict__ in2,
                       float* __restrict__ imp) {
    const int gtid = blockIdx.x * blockDim.x + threadIdx.x;
    const int wave = gtid >> 5;          // 6144 waves total: 3 * 8 * 256
    const int lane = gtid & 31;

    const int hwt = wave & 255;          // 256 tiles of 16 pixels
    const int b   = (wave >> 8) & 7;
    const int t   = wave >> 11;          // 0..2
    const float* in = (t == 0) ? in0 : (t == 1) ? in1 : in2;

    const int hw_base = hwt << 4;
    const int m  = lane & 15;            // A-matrix row (pixel within tile)
    const int kk = (lane >> 4) << 1;     // A-matrix K base: lanes16-31 -> K=2,3
    const float* p = in + ((size_t)b * C_ * HW_) + hw_base + m;

    v8f acc = {0.f, 0.f, 0.f, 0.f, 0.f, 0.f, 0.f, 0.f};
    v2f ones; ones.x = 1.0f; ones.y = 1.0f;    // B = 4x16 all-ones (layout-invariant)

#pragma unroll 4
    for (int c = 0; c < C_; c += 4) {
        v2f a;
        a.x = __builtin_fabsf(p[(size_t)(c + kk)     * HW_]);
        a.y = __builtin_fabsf(p[(size_t)(c + kk + 1) * HW_]);
        // D = A x B + C  (8 args: neg_a, A, neg_b, B, c_mod, C, reuse_a, reuse_b)
        acc = __builtin_amdgcn_wmma_f32_16x16x4_f32(
            false, a, false, ones, (short)0, acc, false, false);
    }

    float* dst = imp + ((size_t)(t * B_ + b) * HW_) + hw_base;
    if (lane == 0) {
#pragma unroll
        for (int j = 0; j < 8; ++j) dst[j] = acc[j];          // rows M=0..7
    } else if (lane == 16) {
#pragma unroll
        for (int j = 0; j < 8; ++j) dst[8 + j] = acc[j];      // rows M=8..15
    }
}

// ---------------------------------------------------------------------------
// Pass 2: per (t,b) pair: k-th largest of 4096 importance values (binary
// search over order-preserving uint keys in LDS), then emit 0/1 mask with
// lax.top_k-compatible stable tie-break (equal values admitted lowest-index
// first). One 256-thread block per pair (24 blocks).
// ---------------------------------------------------------------------------
__device__ __forceinline__ unsigned ordered_key(float f) {
    unsigned u = __float_as_uint(f);
    return u ^ ((u >> 31) ? 0xFFFFFFFFu : 0x80000000u);  // monotonic f32 -> u32
}

__global__ void __launch_bounds__(256)
topk_mask_kernel(const float* __restrict__ imp, float* __restrict__ mask) {
    const int pair = blockIdx.x;     // t*8 + b, 0..23
    const int tid  = threadIdx.x;    // 256 threads

    __shared__ unsigned s_key[HW_];
    __shared__ unsigned s_cnt;
    __shared__ unsigned s_chunk[256];
    __shared__ unsigned s_off[256];

    const float* src = imp + (size_t)pair * HW_;
    for (int i = tid; i < HW_; i += 256) s_key[i] = ordered_key(src[i]);
    __syncthreads();

    // Largest key T with count(key >= T) >= K_SEL  ==  k-th largest value.
    unsigned lo = 0u, hi = 0xFFFFFFFFu;
    while (lo < hi) {
        unsigned mid = lo + ((hi - lo) >> 1) + 1u;
        if (tid == 0) s_cnt = 0u;
        __syncthreads();
        unsigned c = 0u;
        for (int i = tid; i < HW_; i += 256) c += (s_key[i] >= mid) ? 1u : 0u;
        atomicAdd(&s_cnt, c);
        __syncthreads();
        unsigned tot = s_cnt;
        __syncthreads();
        if (tot >= K_SEL) lo = mid; else hi = mid - 1u;
    }
    const unsigned V = lo;

    // Strictly-greater count -> how many ties at V we must admit.
    if (tid == 0) s_cnt = 0u;
    __syncthreads();
    {
        unsigned c = 0u;
        for (int i = tid; i < HW_; i += 256) c += (s_key[i] > V) ? 1u : 0u;
        atomicAdd(&s_cnt, c);
    }
    __syncthreads();
    const unsigned need_eq = K_SEL - s_cnt;
    __syncthreads();

    // Stable (index-order) admission of ties: scan equal-counts per 16-elem chunk.
    const int base_i = tid * 16;
    unsigned eq = 0u;
#pragma unroll
    for (int j = 0; j < 16; ++j) eq += (s_key[base_i + j] == V) ? 1u : 0u;
    s_chunk[tid] = eq;
    __syncthreads();
    if (tid == 0) {
        unsigned run = 0u;
        for (int i = 0; i < 256; ++i) { s_off[i] = run; run += s_chunk[i]; }
    }
    __syncthreads();

    unsigned run = s_off[tid];
    float* dst = mask + (size_t)pair * HW_;
#pragma unroll
    for (int j = 0; j < 16; ++j) {
        const unsigned key = s_key[base_i + j];
        float mval = 0.f;
        if (key > V) {
            mval = 1.f;
        } else if (key == V) {
            if (run < need_eq) mval = 1.f;
            ++run;
        }
        dst[base_i + j] = mval;
    }
}

// ---------------------------------------------------------------------------
// Pass 3: out = fm * mask (broadcast over channels), float4 vectorized.
// Inputs are likely still L2-resident from pass 1 (100MB < 192MB L2).
// ---------------------------------------------------------------------------
__global__ void __launch_bounds__(256)
apply_mask_kernel(const float* __restrict__ in0,
                  const float* __restrict__ in1,
                  const float* __restrict__ in2,
                  const float* __restrict__ mask,
                  float* __restrict__ out) {
    const long long N4 = (long long)B_ * C_ * HW_ / 4;    // 2,097,152 per tensor
    const long long i  = (long long)blockIdx.x * blockDim.x + threadIdx.x;
    if (i >= 3 * N4) return;

    const int t = (int)(i / N4);
    const long long r = i - (long long)t * N4;            // [b][c][hw4], 8x256x1024
    const float* in = (t == 0) ? in0 : (t == 1) ? in1 : in2;

    const int hw4 = (int)(r & 1023);                      // HW/4 = 1024
    const int b   = (int)(r >> 18);                       // r / (256*1024)

    const float4 v  = ((const float4*)in)[r];
    const float4 mk = ((const float4*)mask)[((size_t)(t * B_ + b) << 10) + hw4];
    float4 o;
    o.x = v.x * mk.x; o.y = v.y * mk.y; o.z = v.z * mk.z; o.w = v.w * mk.w;
    ((float4*)out)[i] = o;
}

// ---------------------------------------------------------------------------
extern "C" void kernel_launch(void* const* d_in, const int* in_sizes, int n_in,
                              void* d_out, int out_size, void* d_ws, size_t ws_size,
                              hipStream_t stream) {
    (void)in_sizes; (void)n_in; (void)out_size; (void)ws_size;
    const float* in0 = (const float*)d_in[0];
    const float* in1 = (const float*)d_in[1];
    const float* in2 = (const float*)d_in[2];
    float* out = (float*)d_out;

    float* imp  = (float*)d_ws;                    // 3*8*4096 floats
    float* mask = imp + 3 * B_ * HW_;              // 3*8*4096 floats

    // Pass 1: 3*8*256 = 6144 waves -> 768 blocks of 256 threads (8 waves/block).
    importance_wmma_kernel<<<768, 256, 0, stream>>>(in0, in1, in2, imp);

    // Pass 2: one block per (tensor, batch) pair.
    topk_mask_kernel<<<24, 256, 0, stream>>>(imp, mask);

    // Pass 3: 3 * 2,097,152 float4 elements -> 24576 blocks of 256.
    apply_mask_kernel<<<24576, 256, 0, stream>>>(in0, in1, in2, mask, out);
}